// AGNN_6468220748546
// MI455X (gfx1250) — compile-verified
//
#include <hip/hip_runtime.h>

#define NN 50000
#define EE 400000
#define HH 128
#define NOISE_D 64
#define TT 128
#define GG 16
#define EPSF 1e-5f

typedef __attribute__((ext_vector_type(2))) float v2f;
typedef __attribute__((ext_vector_type(8))) float v8f;

// --------------------------- async LDS staging -----------------------------
typedef __attribute__((address_space(3))) void lds_void;

__device__ __forceinline__ unsigned to_lds_off(void* p) {
  // generic -> LDS address space, then ptrtoint gives byte offset from LDS base
  return (unsigned)(unsigned long long)(lds_void*)p;
}

__device__ __forceinline__ void async_ld_b128(unsigned lds_off,
                                              const void* gaddr) {
  asm volatile("global_load_async_to_lds_b128 %0, %1, off"
               :
               : "v"(lds_off), "v"(gaddr)
               : "memory");
}

__device__ __forceinline__ void wait_async0() {
  asm volatile("s_wait_asynccnt 0x0" ::: "memory");
}

// ---------------------------------------------------------------------------
// fp32 WMMA 16x16 tile: D += A[16xK] * B[Kx16], K multiple of 4.
// A layout (ISA 7.12.2): lane l<16 holds row M=l, VGPR0=K, VGPR1=K+1;
// lanes 16..31 hold K+2/K+3 for the same rows. B: row K striped across lanes.
// ---------------------------------------------------------------------------
__device__ __forceinline__ v8f wmma_f32_tile(const float* __restrict__ A, int lda,
                                             const float* __restrict__ B, int ldb,
                                             int K, v8f c) {
  const int lane = threadIdx.x & 31;
  const int ph   = lane >> 4;   // wave half
  const int l    = lane & 15;
  const float* arow = A + (size_t)l * lda;
  for (int k = 0; k < K; k += 4) {
    v2f a, b;
    a.x = arow[k + 2 * ph];
    a.y = arow[k + 2 * ph + 1];
    b.x = B[(size_t)(k + 2 * ph) * ldb + l];
    b.y = B[(size_t)(k + 2 * ph + 1) * ldb + l];
    c = __builtin_amdgcn_wmma_f32_16x16x4_f32(false, a, false, b, (short)0, c,
                                              false, false);
  }
  return c;
}

// One 16x16 output tile of sx[16x128] @ W[128x128] + bias, per wave.
__device__ __forceinline__ void proj_store(const float* sx,
                                           const float* __restrict__ W,
                                           const float* __restrict__ bias,
                                           float* __restrict__ Y, size_t row0) {
  const int wave = threadIdx.x >> 5, col0 = wave * 16;
  const int lane = threadIdx.x & 31, ph = lane >> 4, l = lane & 15;
  v8f c = {};
  c = wmma_f32_tile(sx, HH, W + col0, HH, HH, c);
  const float bc = bias[col0 + l];
#pragma unroll
  for (int r = 0; r < 8; ++r)
    Y[(row0 + r + 8 * ph) * HH + col0 + l] = c[r] + bc;
}

// Fused node projections: stage features tile once (async->LDS), compute
// Xq/Xr/Xv/Xu = features@W* + b* reusing the same A tile.
__global__ void node_proj4_kernel(const float* __restrict__ X,
                                  const float* __restrict__ Wq,
                                  const float* __restrict__ bq,
                                  float* __restrict__ Yq,
                                  const float* __restrict__ Wr,
                                  const float* __restrict__ br,
                                  float* __restrict__ Yr,
                                  const float* __restrict__ Wv,
                                  const float* __restrict__ bv,
                                  float* __restrict__ Yv,
                                  const float* __restrict__ Wu,
                                  const float* __restrict__ bu,
                                  float* __restrict__ Yu) {
  __shared__ float sx[16 * HH];
  const size_t row0 = (size_t)blockIdx.x * 16;
  const char* gsrc = (const char*)(X + row0 * HH);
  const unsigned lb = to_lds_off(sx);
  const int b = threadIdx.x * 16;          // 256 lanes x 16B = 4KB per issue
  async_ld_b128(lb + b, gsrc + b);
  async_ld_b128(lb + b + 4096, gsrc + b + 4096);
  wait_async0();
  __syncthreads();
  proj_store(sx, Wq, bq, Yq, row0);
  proj_store(sx, Wr, br, Yr, row0);
  proj_store(sx, Wv, bv, Yv, row0);
  proj_store(sx, Wu, bu, Yu, row0);
}

// TE[16,128] = relu(time_emb[16,128]) @ Wt[128,128] + bt   (single block)
__global__ void time_proj_kernel(const float* __restrict__ temb,
                                 const float* __restrict__ Wt,
                                 const float* __restrict__ bt,
                                 float* __restrict__ TE) {
  __shared__ float sa[GG * TT];
  for (int i = threadIdx.x; i < GG * TT; i += blockDim.x)
    sa[i] = fmaxf(temb[i], 0.f);
  __syncthreads();
  const int wave = threadIdx.x >> 5, col0 = wave * 16;
  const int lane = threadIdx.x & 31, ph = lane >> 4, l = lane & 15;
  v8f c = {};
  c = wmma_f32_tile(sa, TT, Wt + col0, HH, TT, c);
  const float bc = bt[col0 + l];
#pragma unroll
  for (int r = 0; r < 8; ++r)
    TE[(size_t)(r + 8 * ph) * HH + col0 + l] = c[r] + bc;
}

// Fused edge stage: nme = noise@We+be ; e_hat = nme@Wp+bp+Xq[src]+Xr[dst];
// gate = sigmoid(e_hat); aggr[src] += Xv[dst]*gate
__global__ void edge_stage_kernel(const float* __restrict__ noise,
                                  const float* __restrict__ We,
                                  const float* __restrict__ be,
                                  const float* __restrict__ Wp,
                                  const float* __restrict__ bp,
                                  const int* __restrict__ eidx,
                                  const float* __restrict__ Xq,
                                  const float* __restrict__ Xr,
                                  const float* __restrict__ Xv,
                                  float* __restrict__ nme,
                                  float* __restrict__ ebuf,
                                  float* __restrict__ aggr) {
  __shared__ float snoise[16 * NOISE_D];
  __shared__ float snme[16 * HH];
  __shared__ int ssrc[16], sdst[16];
  const int e0 = blockIdx.x * 16;
  const int wave = threadIdx.x >> 5, col0 = wave * 16;
  const int lane = threadIdx.x & 31, ph = lane >> 4, l = lane & 15;

  // stage the shared 16x64 noise tile once per block (4KB = one b128/lane)
  const char* gnoise = (const char*)(noise + (size_t)e0 * NOISE_D);
  const unsigned lb = to_lds_off(snoise);
  const int b = threadIdx.x * 16;
  async_ld_b128(lb + b, gnoise + b);
  if (threadIdx.x < 16) {
    ssrc[threadIdx.x] = eidx[e0 + threadIdx.x];
    sdst[threadIdx.x] = eidx[EE + e0 + threadIdx.x];
  }
  wait_async0();
  __syncthreads();

  v8f c = {};
  c = wmma_f32_tile(snoise, NOISE_D, We + col0, HH, NOISE_D, c);
  const float bec = be[col0 + l];
#pragma unroll
  for (int r = 0; r < 8; ++r) {
    const int rl = r + 8 * ph;
    const float v = c[r] + bec;
    snme[rl * HH + col0 + l] = v;
    nme[(size_t)(e0 + rl) * HH + col0 + l] = v;
  }
  __syncthreads();
  v8f c2 = {};
  c2 = wmma_f32_tile(snme, HH, Wp + col0, HH, HH, c2);
  const float bpc = bp[col0 + l];
#pragma unroll
  for (int r = 0; r < 8; ++r) {
    const int rl = r + 8 * ph;
    const int s = ssrc[rl], d = sdst[rl];
    const float eh = c2[r] + bpc + Xq[(size_t)s * HH + col0 + l] +
                     Xr[(size_t)d * HH + col0 + l];
    ebuf[(size_t)(e0 + rl) * HH + col0 + l] = eh;
    const float g = 1.f / (1.f + __expf(-eh));
    atomicAdd(&aggr[(size_t)s * HH + col0 + l],
              Xv[(size_t)d * HH + col0 + l] * g);
  }
}

// ------------------------- GraphNorm (nodes) -------------------------------
__global__ void gn_node_a_kernel(const float* __restrict__ Xu,
                                 const float* __restrict__ aggr,
                                 const int* __restrict__ batch,
                                 float* __restrict__ hbuf,
                                 float* __restrict__ sum,
                                 float* __restrict__ cnt) {
  const int total = NN * HH;
  for (int i = blockIdx.x * blockDim.x + threadIdx.x; i < total;
       i += gridDim.x * blockDim.x) {
    const int n = i >> 7, ch = i & 127;
    const float v = Xu[i] + aggr[i];
    hbuf[i] = v;
    const int g = batch[n];
    atomicAdd(&sum[g * HH + ch], v);
    if (ch == 0) atomicAdd(&cnt[g], 1.f);
  }
}

__global__ void gn_node_b_kernel(const int* __restrict__ batch,
                                 const float* __restrict__ ms,
                                 const float* __restrict__ sum,
                                 const float* __restrict__ cnt,
                                 float* __restrict__ hbuf,
                                 float* __restrict__ sumsq) {
  const int total = NN * HH;
  for (int i = blockIdx.x * blockDim.x + threadIdx.x; i < total;
       i += gridDim.x * blockDim.x) {
    const int n = i >> 7, ch = i & 127;
    const int g = batch[n];
    const float cn = fmaxf(cnt[g], 1.f);
    const float o = hbuf[i] - (sum[g * HH + ch] / cn) * ms[ch];
    hbuf[i] = o;
    atomicAdd(&sumsq[g * HH + ch], o * o);
  }
}

__global__ void gn_node_c_kernel(const int* __restrict__ batch,
                                 const float* __restrict__ w,
                                 const float* __restrict__ b,
                                 const float* __restrict__ sumsq,
                                 const float* __restrict__ cnt,
                                 const float* __restrict__ features,
                                 const float* __restrict__ hbuf,
                                 float* __restrict__ hout) {
  const int total = NN * HH;
  for (int i = blockIdx.x * blockDim.x + threadIdx.x; i < total;
       i += gridDim.x * blockDim.x) {
    const int n = i >> 7, ch = i & 127;
    const int g = batch[n];
    const float cn = fmaxf(cnt[g], 1.f);
    const float var = sumsq[g * HH + ch] / cn;
    const float hn = w[ch] * hbuf[i] * rsqrtf(var + EPSF) + b[ch];
    hout[i] = features[i] + fmaxf(hn, 0.f);
  }
}

// ------------------------- GraphNorm (edges) -------------------------------
__global__ void gn_edge_a_kernel(const int* __restrict__ eidx,
                                 const int* __restrict__ batch,
                                 const float* __restrict__ ebuf,
                                 float* __restrict__ sum,
                                 float* __restrict__ cnt) {
  const int total = EE * HH;
  for (int i = blockIdx.x * blockDim.x + threadIdx.x; i < total;
       i += gridDim.x * blockDim.x) {
    const int e = i >> 7, ch = i & 127;
    const int g = batch[eidx[e]];
    atomicAdd(&sum[g * HH + ch], ebuf[i]);
    if (ch == 0) atomicAdd(&cnt[g], 1.f);
  }
}

__global__ void gn_edge_b_kernel(const int* __restrict__ eidx,
                                 const int* __restrict__ batch,
                                 const float* __restrict__ ms,
                                 const float* __restrict__ sum,
                                 const float* __restrict__ cnt,
                                 float* __restrict__ ebuf,
                                 float* __restrict__ sumsq) {
  const int total = EE * HH;
  for (int i = blockIdx.x * blockDim.x + threadIdx.x; i < total;
       i += gridDim.x * blockDim.x) {
    const int e = i >> 7, ch = i & 127;
    const int g = batch[eidx[e]];
    const float cn = fmaxf(cnt[g], 1.f);
    const float o = ebuf[i] - (sum[g * HH + ch] / cn) * ms[ch];
    ebuf[i] = o;
    atomicAdd(&sumsq[g * HH + ch], o * o);
  }
}

__global__ void gn_edge_c_kernel(const int* __restrict__ eidx,
                                 const int* __restrict__ batch,
                                 const float* __restrict__ w,
                                 const float* __restrict__ b,
                                 const float* __restrict__ sumsq,
                                 const float* __restrict__ cnt,
                                 const float* __restrict__ TE,
                                 float* __restrict__ ebuf) {
  const int total = EE * HH;
  for (int i = blockIdx.x * blockDim.x + threadIdx.x; i < total;
       i += gridDim.x * blockDim.x) {
    const int e = i >> 7, ch = i & 127;
    const int g = batch[eidx[e]];
    const float cn = fmaxf(cnt[g], 1.f);
    const float var = sumsq[g * HH + ch] / cn;
    const float en = w[ch] * ebuf[i] * rsqrtf(var + EPSF) + b[ch];
    ebuf[i] = fmaxf(en, 0.f) + TE[g * HH + ch];
  }
}

// Final: per-row LayerNorm -> SiLU (in LDS), then e = nme + silu(ln)@Wo + bo
__global__ void final_edge_kernel(const float* __restrict__ Wo,
                                  const float* __restrict__ bo,
                                  const float* __restrict__ lnw,
                                  const float* __restrict__ lnb,
                                  const float* __restrict__ nme,
                                  float* __restrict__ ebuf) {
  __shared__ float s[16 * HH];
  const size_t base = (size_t)blockIdx.x * 16 * HH;
  // async-stage the 16x128 e tile (8KB = 2 b128 issues per lane)
  const char* gsrc = (const char*)(ebuf + base);
  const unsigned lb = to_lds_off(s);
  const int bb = threadIdx.x * 16;
  async_ld_b128(lb + bb, gsrc + bb);
  async_ld_b128(lb + bb + 4096, gsrc + bb + 4096);
  wait_async0();
  __syncthreads();
  const int wave = threadIdx.x >> 5;
  const int lane = threadIdx.x & 31;
#pragma unroll
  for (int rr = 0; rr < 2; ++rr) {       // 8 waves x 2 rows = 16 rows
    float* rp = s + (wave * 2 + rr) * HH;
    float acc = 0.f;
    for (int j = lane; j < HH; j += 32) acc += rp[j];
    for (int o = 16; o > 0; o >>= 1) acc += __shfl_xor(acc, o, 32);
    const float mu = acc * (1.f / HH);
    float vs = 0.f;
    for (int j = lane; j < HH; j += 32) { float d = rp[j] - mu; vs += d * d; }
    for (int o = 16; o > 0; o >>= 1) vs += __shfl_xor(vs, o, 32);
    const float inv = rsqrtf(vs * (1.f / HH) + EPSF);
    for (int j = lane; j < HH; j += 32) {
      const float ln = (rp[j] - mu) * inv * lnw[j] + lnb[j];
      rp[j] = ln / (1.f + __expf(-ln));   // SiLU
    }
  }
  __syncthreads();
  const int col0 = wave * 16, ph = lane >> 4, l = lane & 15;
  v8f c = {};
  c = wmma_f32_tile(s, HH, Wo + col0, HH, HH, c);
  const float bc = bo[col0 + l];
#pragma unroll
  for (int r = 0; r < 8; ++r) {
    const size_t o = base + (size_t)(r + 8 * ph) * HH + col0 + l;
    ebuf[o] = nme[o] + c[r] + bc;
  }
}

extern "C" void kernel_launch(void* const* d_in, const int* in_sizes, int n_in,
                              void* d_out, int out_size, void* d_ws,
                              size_t ws_size, hipStream_t stream) {
  (void)in_sizes; (void)n_in; (void)out_size; (void)ws_size;
  const float* features = (const float*)d_in[0];
  const int*   eidx     = (const int*)d_in[1];
  const float* noise    = (const float*)d_in[2];
  const float* temb     = (const float*)d_in[3];
  const int*   batch    = (const int*)d_in[4];
  const float* We = (const float*)d_in[5];  const float* be = (const float*)d_in[6];
  const float* Wp = (const float*)d_in[7];  const float* bp = (const float*)d_in[8];
  const float* Wq = (const float*)d_in[9];  const float* bq = (const float*)d_in[10];
  const float* Wr = (const float*)d_in[11]; const float* br = (const float*)d_in[12];
  const float* Wu = (const float*)d_in[13]; const float* bu = (const float*)d_in[14];
  const float* Wv = (const float*)d_in[15]; const float* bv = (const float*)d_in[16];
  const float* gnh_w  = (const float*)d_in[17];
  const float* gnh_b  = (const float*)d_in[18];
  const float* gnh_ms = (const float*)d_in[19];
  const float* gne_w  = (const float*)d_in[20];
  const float* gne_b  = (const float*)d_in[21];
  const float* gne_ms = (const float*)d_in[22];
  const float* Wt  = (const float*)d_in[23]; const float* bt = (const float*)d_in[24];
  const float* lnw = (const float*)d_in[25]; const float* lnb = (const float*)d_in[26];
  const float* Wo  = (const float*)d_in[27]; const float* bo = (const float*)d_in[28];

  float* ws    = (float*)d_ws;
  float* Xq    = ws;
  float* Xr    = Xq + (size_t)NN * HH;
  float* Xv    = Xr + (size_t)NN * HH;
  float* Xu    = Xv + (size_t)NN * HH;
  float* aggr  = Xu + (size_t)NN * HH;
  float* nme   = aggr + (size_t)NN * HH;
  float* stats = nme + (size_t)EE * HH;
  float* sum_h   = stats;
  float* sumsq_h = sum_h + GG * HH;
  float* sum_e   = sumsq_h + GG * HH;
  float* sumsq_e = sum_e + GG * HH;
  float* cnt_h   = sumsq_e + GG * HH;
  float* cnt_e   = cnt_h + GG;
  float* TE      = cnt_e + GG;
  float* hbuf    = Xq;  // Xq is dead after edge_stage; reuse for GraphNorm(h)

  float* hout = (float*)d_out;
  float* ebuf = hout + (size_t)NN * HH;  // e half of d_out doubles as scratch

  hipMemsetAsync(aggr, 0, (size_t)NN * HH * sizeof(float), stream);
  hipMemsetAsync(stats, 0, (size_t)(4 * GG * HH + 2 * GG) * sizeof(float),
                 stream);

  node_proj4_kernel<<<NN / 16, 256, 0, stream>>>(features, Wq, bq, Xq, Wr, br,
                                                 Xr, Wv, bv, Xv, Wu, bu, Xu);
  time_proj_kernel<<<1, 256, 0, stream>>>(temb, Wt, bt, TE);

  edge_stage_kernel<<<EE / 16, 256, 0, stream>>>(noise, We, be, Wp, bp, eidx,
                                                 Xq, Xr, Xv, nme, ebuf, aggr);

  const int EW = 2048;
  gn_node_a_kernel<<<EW, 256, 0, stream>>>(Xu, aggr, batch, hbuf, sum_h, cnt_h);
  gn_node_b_kernel<<<EW, 256, 0, stream>>>(batch, gnh_ms, sum_h, cnt_h, hbuf,
                                           sumsq_h);
  gn_node_c_kernel<<<EW, 256, 0, stream>>>(batch, gnh_w, gnh_b, sumsq_h, cnt_h,
                                           features, hbuf, hout);

  gn_edge_a_kernel<<<EW, 256, 0, stream>>>(eidx, batch, ebuf, sum_e, cnt_e);
  gn_edge_b_kernel<<<EW, 256, 0, stream>>>(eidx, batch, gne_ms, sum_e, cnt_e,
                                           ebuf, sumsq_e);
  gn_edge_c_kernel<<<EW, 256, 0, stream>>>(eidx, batch, gne_w, gne_b, sumsq_e,
                                           cnt_e, TE, ebuf);

  final_edge_kernel<<<EE / 16, 256, 0, stream>>>(Wo, bo, lnw, lnb, nme, ebuf);
}